// ReIdHead_33234456937214
// MI455X (gfx1250) — compile-verified
//
#include <hip/hip_runtime.h>
#include <hip/hip_bf16.h>

typedef __attribute__((ext_vector_type(2))) float v2f;
typedef __attribute__((ext_vector_type(8))) float v8f;

#define DDIM 128
#define NQ   64
#define EPSF 1e-8f

// ---------------------------------------------------------------------------
// Kernel 1: L2-normalize the 64 queries (64 blocks x 128 threads)
// ---------------------------------------------------------------------------
__global__ void qnorm_kernel(const float* __restrict__ q, float* __restrict__ qn) {
    const int m    = blockIdx.x;
    const int t    = threadIdx.x;          // 0..127
    const int lane = t & 31;
    const int wid  = t >> 5;               // 4 waves
    __shared__ float red[4];

    float x  = q[m * DDIM + t];
    float ss = x * x;
    ss += __shfl_xor(ss, 16, 32);
    ss += __shfl_xor(ss,  8, 32);
    ss += __shfl_xor(ss,  4, 32);
    ss += __shfl_xor(ss,  2, 32);
    ss += __shfl_xor(ss,  1, 32);
    if (lane == 0) red[wid] = ss;
    __syncthreads();
    float tot = red[0] + red[1] + red[2] + red[3];
    float inv = 1.0f / fmaxf(sqrtf(tot), EPSF);
    qn[m * DDIM + t] = x * inv;
}

// ---------------------------------------------------------------------------
// Kernel 2: streaming WMMA similarity + per-wave running argmax
//   block = 256 threads = 8 waves; each wave owns 16 DB rows per group.
//   C tile = 16 queries x 16 rows; K accumulated 4 at a time via
//   v_wmma_f32_16x16x4_f32; 4 query-tiles cover all 64 queries.
//   K-loop is only partially unrolled to keep A fragments flowing from LDS
//   (ds_load_b64) instead of being hoisted+spilled to scratch.
// ---------------------------------------------------------------------------
__global__ void nn_main_kernel(const float* __restrict__ db,
                               const float* __restrict__ qn_g,
                               float* __restrict__ candVal,
                               int*   __restrict__ candIdx,
                               int numGroups, int nRows, int totalWaves) {
    __shared__ float qn[NQ * DDIM];        // 32 KB
    for (int i = threadIdx.x; i < NQ * DDIM; i += blockDim.x) qn[i] = qn_g[i];
    __syncthreads();

    const int lane  = threadIdx.x & 31;
    const int wave  = threadIdx.x >> 5;
    const int gwave = blockIdx.x * 8 + wave;
    const int nlo   = lane & 15;
    const bool hi   = lane >= 16;
    const int ksh   = hi ? 2 : 0;

    // Per-tile LDS base pointers for this lane's A fragments.
    const float* qp0 = &qn[(0 * 16 + nlo) * DDIM + ksh];
    const float* qp1 = &qn[(1 * 16 + nlo) * DDIM + ksh];
    const float* qp2 = &qn[(2 * 16 + nlo) * DDIM + ksh];
    const float* qp3 = &qn[(3 * 16 + nlo) * DDIM + ksh];

    float bestv[4][8];
    int   besti[4][8];
#pragma unroll
    for (int tq = 0; tq < 4; ++tq)
#pragma unroll
        for (int r = 0; r < 8; ++r) { bestv[tq][r] = -3.4e38f; besti[tq][r] = 0; }

    for (int g = gwave; g < numGroups; g += totalWaves) {
        int row  = g * 16 + nlo;
        int rowc = row < nRows ? row : (nRows - 1);
        const float* rowp = db + (size_t)rowc * DDIM;

        v8f acc0 = (v8f)0.0f, acc1 = (v8f)0.0f, acc2 = (v8f)0.0f, acc3 = (v8f)0.0f;
        float ss = 0.0f;

#pragma unroll 4
        for (int k = 0; k < DDIM; k += 4) {
            const float4 v = *(const float4*)(rowp + k);
            ss += v.x * v.x + v.y * v.y + v.z * v.z + v.w * v.w;

            v2f b;
            b.x = hi ? v.z : v.x;     // B fragment: K = k+ksh, k+ksh+1
            b.y = hi ? v.w : v.y;

            v2f a0 = *(const v2f*)(qp0 + k);   // ds_load_b64 each iteration
            v2f a1 = *(const v2f*)(qp1 + k);
            v2f a2 = *(const v2f*)(qp2 + k);
            v2f a3 = *(const v2f*)(qp3 + k);

            acc0 = __builtin_amdgcn_wmma_f32_16x16x4_f32(false, a0, false, b, (short)0, acc0, false, false);
            acc1 = __builtin_amdgcn_wmma_f32_16x16x4_f32(false, a1, false, b, (short)0, acc1, false, false);
            acc2 = __builtin_amdgcn_wmma_f32_16x16x4_f32(false, a2, false, b, (short)0, acc2, false, false);
            acc3 = __builtin_amdgcn_wmma_f32_16x16x4_f32(false, a3, false, b, (short)0, acc3, false, false);
        }

        const float inv   = 1.0f / fmaxf(sqrtf(ss), EPSF);
        const bool  valid = row < nRows;
#pragma unroll
        for (int r = 0; r < 8; ++r) {
            float s0 = acc0[r] * inv;
            float s1 = acc1[r] * inv;
            float s2 = acc2[r] * inv;
            float s3 = acc3[r] * inv;
            if (valid && s0 > bestv[0][r]) { bestv[0][r] = s0; besti[0][r] = row; }
            if (valid && s1 > bestv[1][r]) { bestv[1][r] = s1; besti[1][r] = row; }
            if (valid && s2 > bestv[2][r]) { bestv[2][r] = s2; besti[2][r] = row; }
            if (valid && s3 > bestv[3][r]) { bestv[3][r] = s3; besti[3][r] = row; }
        }
    }

    // Cross-lane argmax over the 16-lane halves (rows), then emit candidates.
#pragma unroll
    for (int tq = 0; tq < 4; ++tq)
#pragma unroll
        for (int r = 0; r < 8; ++r) {
            float v = bestv[tq][r];
            int   i = besti[tq][r];
#pragma unroll
            for (int m = 8; m >= 1; m >>= 1) {
                float ov = __shfl_xor(v, m, 32);
                int   oi = __shfl_xor(i, m, 32);
                if (ov > v || (ov == v && oi < i)) { v = ov; i = oi; }
            }
            if (lane == 0) {            // queries tq*16 + r      (from lanes 0..15)
                int qidx = tq * 16 + r;
                candVal[(size_t)gwave * NQ + qidx] = v;
                candIdx[(size_t)gwave * NQ + qidx] = i;
            }
            if (lane == 16) {           // queries tq*16 + 8 + r  (from lanes 16..31)
                int qidx = tq * 16 + 8 + r;
                candVal[(size_t)gwave * NQ + qidx] = v;
                candIdx[(size_t)gwave * NQ + qidx] = i;
            }
        }
}

// ---------------------------------------------------------------------------
// Kernel 3: final reduction across waves; gather labels; write outputs.
// grid = 64 (one block per query), block = 256
// ---------------------------------------------------------------------------
__global__ void nn_reduce_kernel(const float* __restrict__ candVal,
                                 const int*   __restrict__ candIdx,
                                 const int*   __restrict__ labels,
                                 int W, char* __restrict__ out, int i64mode) {
    const int q   = blockIdx.x;
    const int tid = threadIdx.x;
    __shared__ float sv[256];
    __shared__ int   si[256];

    float bv = -3.4e38f;
    int   bi = 0;
    for (int w = tid; w < W; w += 256) {
        float v = candVal[(size_t)w * NQ + q];
        int   i = candIdx[(size_t)w * NQ + q];
        if (v > bv || (v == bv && i < bi)) { bv = v; bi = i; }
    }
    sv[tid] = bv; si[tid] = bi;
    __syncthreads();
    for (int s = 128; s > 0; s >>= 1) {
        if (tid < s) {
            float ov = sv[tid + s]; int oi = si[tid + s];
            if (ov > sv[tid] || (ov == sv[tid] && oi < si[tid])) { sv[tid] = ov; si[tid] = oi; }
        }
        __syncthreads();
    }
    if (tid == 0) {
        int lbl = labels[si[0]];
        if (i64mode) {
            ((long long*)out)[q]                        = (long long)lbl;
            ((float*)(out + NQ * sizeof(long long)))[q] = sv[0];
        } else {
            ((float*)out)[q]      = (float)lbl;
            ((float*)out)[NQ + q] = sv[0];
        }
    }
}

// ---------------------------------------------------------------------------
extern "C" void kernel_launch(void* const* d_in, const int* in_sizes, int n_in,
                              void* d_out, int out_size, void* d_ws, size_t ws_size,
                              hipStream_t stream) {
    const float* q      = (const float*)d_in[0];   // [64,128] f32
    const float* db     = (const float*)d_in[1];   // [N,128]  f32
    const int*   labels = (const int*)d_in[2];     // [N]

    const int nRows     = in_sizes[1] / DDIM;
    const int numGroups = (nRows + 15) / 16;

    // Workspace layout: qn (32KB) | candVal (W*64 f32) | candIdx (W*64 i32)
    float* qn = (float*)d_ws;
    size_t fixed   = (size_t)NQ * DDIM * sizeof(float);
    size_t perWave = (size_t)NQ * (sizeof(float) + sizeof(int));
    int maxW = (ws_size > fixed) ? (int)((ws_size - fixed) / perWave) : 8;
    int W = maxW < 8192 ? (maxW / 8) * 8 : 8192;   // multiple of 8 waves/WG
    if (W < 8) W = 8;
    int numWG = W / 8;

    float* candVal = qn + (size_t)NQ * DDIM;
    int*   candIdx = (int*)(candVal + (size_t)W * NQ);

    int i64mode = (out_size * (int)sizeof(float) >=
                   NQ * (int)sizeof(long long) + NQ * (int)sizeof(float)) ? 1 : 0;

    qnorm_kernel<<<NQ, DDIM, 0, stream>>>(q, qn);
    nn_main_kernel<<<numWG, 256, 0, stream>>>(db, qn, candVal, candIdx,
                                              numGroups, nRows, W);
    nn_reduce_kernel<<<NQ, 256, 0, stream>>>(candVal, candIdx, labels,
                                             W, (char*)d_out, i64mode);
}